// SLP_GCN_4node_64862596104506
// MI455X (gfx1250) — compile-verified
//
#include <hip/hip_runtime.h>
#include <hip/hip_bf16.h>
#include <cstdint>
#include <cstddef>

typedef __attribute__((ext_vector_type(2))) float v2f;
typedef __attribute__((ext_vector_type(8))) float v8f;

#define WAVES_PER_BLOCK 8
#define TPB 256

// ---------------- degree + norm ----------------
__global__ void deg_kernel(const int* __restrict__ src, const int* __restrict__ dst,
                           float* __restrict__ outdeg, float* __restrict__ indeg, int E) {
    int t = blockIdx.x * blockDim.x + threadIdx.x;
    if (t >= E) return;
    atomicAdd(&outdeg[src[t]], 1.0f);
    atomicAdd(&indeg[dst[t]], 1.0f);
}

__global__ void norm_kernel(float* __restrict__ outn, float* __restrict__ inn, int n) {
    int t = blockIdx.x * blockDim.x + threadIdx.x;
    if (t >= n) return;
    outn[t] = rsqrtf(fmaxf(outn[t], 1.0f));
    inn[t]  = rsqrtf(fmaxf(inn[t], 1.0f));
}

// ---------------- edge scatter (128-wide rows) ----------------
// One wave32 per edge; lane l handles float4 chunk l (32*4 = 128 floats).
__global__ void scatter128(const float* __restrict__ g, const int* __restrict__ src,
                           const int* __restrict__ dst, const float* __restrict__ outn,
                           float* __restrict__ agg, int E) {
    long long t = (long long)blockIdx.x * blockDim.x + threadIdx.x;
    int e = (int)(t >> 5);
    int lane = (int)(t & 31);
    if (e >= E) return;
    int s = src[e];
    int d = dst[e];
    float sc = outn[s];
    float4 v = ((const float4*)(g + (size_t)s * 128))[lane];
    float* ap = agg + (size_t)d * 128 + lane * 4;
    atomicAdd(ap + 0, v.x * sc);
    atomicAdd(ap + 1, v.y * sc);
    atomicAdd(ap + 2, v.z * sc);
    atomicAdd(ap + 3, v.w * sc);
}

// ---------------- elementwise finish: out = relu(agg*inn + bias), N = 128 ----------------
__global__ void finish128(const float* __restrict__ agg, const float* __restrict__ inn,
                          const float* __restrict__ bias, float* __restrict__ out,
                          int M, int relu) {
    int t = blockIdx.x * blockDim.x + threadIdx.x;     // over M*32 float4s
    if (t >= M * 32) return;
    int row = t >> 5;
    int c4  = (t & 31) << 2;
    float s = inn[row];
    float4 v = ((const float4*)agg)[t];
    float4 o;
    o.x = v.x * s + bias[c4 + 0];
    o.y = v.y * s + bias[c4 + 1];
    o.z = v.z * s + bias[c4 + 2];
    o.w = v.w * s + bias[c4 + 3];
    if (relu) {
        o.x = fmaxf(o.x, 0.0f); o.y = fmaxf(o.y, 0.0f);
        o.z = fmaxf(o.z, 0.0f); o.w = fmaxf(o.w, 0.0f);
    }
    ((float4*)out)[t] = o;
}

// ---------------- WMMA f32 GEMM:  out[M,N] = (rowscale.*A)[M,K] @ W[K,N] (+bias)(+relu) ----
// One wave32 per 16-row strip of A; W chunks of KC<=128 rows staged in LDS in a
// PAIRED-ROW layout: pair p stores {W[2p][c], W[2p+1][c]} as one float2, so each
// lane's B fragment is a single aligned ds_load_b64 (no repack movs).
// Pair-row stride padded to N+16 float2s: per-row dword shift 2(N+16) == 32 (mod 64),
// so half-waves (pair p vs p+1) hit disjoint bank halves -> conflict-free.
// A fragment (V_WMMA_F32_16X16X4_F32, ISA 7.12.2):
//   lane 0-15: M=lane, {K0,K1}; lane 16-31: M=lane-16, {K2,K3}
// C/D layout: VGPR v -> M = v + 8*(lane>=16), N = lane%16.
template <int NT>
__global__ void gemm_wmma_f32(const float* __restrict__ A, const float* __restrict__ W,
                              const float* __restrict__ bias, const float* __restrict__ rowscale,
                              float* __restrict__ out, int M, int K, int relu) {
    constexpr int N  = NT * 16;
    constexpr int PS = N + 16;               // padded pair-row stride (float2 units)
    extern __shared__ float lds[];
    float2* lds2 = (float2*)lds;

    const int lane  = threadIdx.x & 31;
    const int wave  = threadIdx.x >> 5;
    const int strip = blockIdx.x * WAVES_PER_BLOCK + wave;
    const bool active = (strip * 16) < M;
    const int row0  = strip * 16;
    const int mlane = lane & 15;
    const int koff  = (lane >> 4) << 1;      // 0 for lanes 0-15, 2 for lanes 16-31

    int arow = row0 + mlane;
    if (arow >= M) arow = M - 1;
    const float* __restrict__ Arow = A + (size_t)arow * K;
    const float scale = (rowscale != nullptr) ? rowscale[arow] : 1.0f;

    v8f acc[NT];
#pragma unroll
    for (int nt = 0; nt < NT; ++nt) acc[nt] = (v8f){};

    const int KC = (K < 128) ? K : 128;
    for (int k0 = 0; k0 < K; k0 += KC) {
        // cooperative load of W chunk [k0 .. k0+KC) x N into paired LDS layout
        const int npair = (KC >> 1) * N;
        for (int i = threadIdx.x; i < npair; i += blockDim.x) {
            int p = i / N;
            int c = i - p * N;
            const float* wp = W + (size_t)(k0 + 2 * p) * N + c;
            float2 pr;
            pr.x = wp[0];
            pr.y = wp[N];
            lds2[p * PS + c] = pr;
        }
        __syncthreads();

        if (active) {
            for (int k = 0; k < KC; k += 4) {
                float2 av = *(const float2*)(Arow + k0 + k + koff);
                v2f a;
                a.x = av.x * scale;
                a.y = av.y * scale;
                const float2* bp = lds2 + ((k + koff) >> 1) * PS + mlane;
#pragma unroll
                for (int nt = 0; nt < NT; ++nt) {
                    float2 bv = bp[nt * 16];       // single ds_load_b64
                    v2f b;
                    b.x = bv.x;
                    b.y = bv.y;
                    acc[nt] = __builtin_amdgcn_wmma_f32_16x16x4_f32(
                        false, a, false, b, (short)0, acc[nt], false, false);
                }
            }
        }
        __syncthreads();
    }

    if (!active) return;

    const int rbase = row0 + ((lane >> 4) << 3);
#pragma unroll
    for (int nt = 0; nt < NT; ++nt) {
        const int col = nt * 16 + mlane;
        const float bv = (bias != nullptr) ? bias[col] : 0.0f;
#pragma unroll
        for (int v = 0; v < 8; ++v) {
            int r = rbase + v;
            if (r < M) {
                float val = acc[nt][v] + bv;
                if (relu) val = fmaxf(val, 0.0f);
                out[(size_t)r * N + col] = val;
            }
        }
    }
}

extern "C" void kernel_launch(void* const* d_in, const int* in_sizes, int n_in,
                              void* d_out, int out_size, void* d_ws, size_t ws_size,
                              hipStream_t stream) {
    const float* x   = (const float*)d_in[0];
    const int*   src = (const int*)d_in[1];
    const int*   dst = (const int*)d_in[2];
    const float* W1  = (const float*)d_in[3];
    const float* b1  = (const float*)d_in[4];
    const float* W2  = (const float*)d_in[5];
    const float* b2  = (const float*)d_in[6];
    const float* W3  = (const float*)d_in[7];
    const float* b3  = (const float*)d_in[8];

    const int n = in_sizes[0] / 256;   // 50000 nodes
    const int E = in_sizes[1];         // 800000 edges
    const int D = 128;

    // workspace layout (floats): out_norm[n] | in_norm[n] | bufP[n*128] | bufQ[n*128]
    float* outn = (float*)d_ws;
    float* inn  = outn + n;
    float* bufP = inn + n;
    float* bufQ = bufP + (size_t)n * D;

    // output layout: h4[n*64] | h3[n*128] | h2[n*128]
    float* h4 = (float*)d_out;
    float* h3 = h4 + (size_t)n * 64;
    float* h2 = h3 + (size_t)n * D;

    const int strips  = (n + 15) / 16;
    const int gblocks = (strips + WAVES_PER_BLOCK - 1) / WAVES_PER_BLOCK;
    const int eblocks = (int)(((long long)E * 32 + TPB - 1) / TPB);
    const int fblocks = (n * 32 + TPB - 1) / TPB;
    const size_t rowbytes = (size_t)n * D * sizeof(float);

    // LDS bytes: (KC/2) pair-rows of stride (N+16) float2s
    const size_t lds8 = (size_t)64 * (128 + 16) * sizeof(float2);  // 73,728 B
    const size_t lds4 = (size_t)64 * (64 + 16) * sizeof(float2);   // 40,960 B

    // ---- degrees -> norms ----
    hipMemsetAsync(outn, 0, sizeof(float) * 2 * (size_t)n, stream);
    deg_kernel<<<(E + TPB - 1) / TPB, TPB, 0, stream>>>(src, dst, outn, inn, E);
    norm_kernel<<<(n + TPB - 1) / TPB, TPB, 0, stream>>>(outn, inn, n);

    // ---- Layer 1 (GEMM-first; linear commutes with node-axis scalings/aggregation):
    // g1 = x @ W1 ; agg = A·D_out·g1 ; h1 = relu(D_in·agg + b1)
    gemm_wmma_f32<8><<<gblocks, TPB, lds8, stream>>>(
        x, W1, nullptr, nullptr, bufP, n, 256, 0);
    hipMemsetAsync(bufQ, 0, rowbytes, stream);
    scatter128<<<eblocks, TPB, 0, stream>>>(bufP, src, dst, outn, bufQ, E);
    finish128<<<fblocks, TPB, 0, stream>>>(bufQ, inn, b1, bufP, n, 1);   // h1 -> bufP

    // ---- Layer 2: g2 = h1 @ W2 ; h2 = relu(D_in·A·D_out·g2 + b2) -> d_out
    gemm_wmma_f32<8><<<gblocks, TPB, lds8, stream>>>(
        bufP, W2, nullptr, nullptr, bufQ, n, 128, 0);
    hipMemsetAsync(bufP, 0, rowbytes, stream);
    scatter128<<<eblocks, TPB, 0, stream>>>(bufQ, src, dst, outn, bufP, E);
    finish128<<<fblocks, TPB, 0, stream>>>(bufP, inn, b2, h2, n, 1);     // h2 -> d_out

    // ---- Shared aggregation of h2 for layers 3 and 4 (aggregate-first):
    hipMemsetAsync(bufQ, 0, rowbytes, stream);
    scatter128<<<eblocks, TPB, 0, stream>>>(h2, src, dst, outn, bufQ, E);

    // ---- Layer 3: h3 = relu((D_in·agg) @ W2 + b2)  (rowscale folded into A-load)
    gemm_wmma_f32<8><<<gblocks, TPB, lds8, stream>>>(
        bufQ, W2, b2, inn, h3, n, 128, 1);

    // ---- Layer 4: h4 = (D_in·agg) @ W3 + b3
    gemm_wmma_f32<4><<<gblocks, TPB, lds4, stream>>>(
        bufQ, W3, b3, inn, h4, n, 128, 0);
}